// MPJPE_Loss_74912819577249
// MI455X (gfx1250) — compile-verified
//
#include <hip/hip_runtime.h>
#include <math.h>

// ---------------------------------------------------------------------------
// MPJPE-style loss: l3D + 0.05 * (L2D + LHEM)
//   l3D  = sum |joint3d - pred|                       (672 elems)
//   L2D  = sum (middle_out[0,:17] - heatmap)^2        (17 x 65536)
//   LHEM = sum_p ( sum_{j<3} ||T_GT[p,j]-T_pred[p,j]||_F )^2
//
// Memory-bound: ~20 MB HBM traffic @ 23.3 TB/s ~ 0.9 us.
//   - middle_out[0,0:59] (15.5 MB) read once  -> non-temporal b128 loads
//   - heatmap (4.5 MB) reused by 42 T-tasks   -> regular loads, stays in L2
// WMMA f32 16x16x4 (all-ones B => row sums in D) is the wave-32 reduction
// primitive; cross-half combine is a single ds_swizzle SWAPX16.
// ---------------------------------------------------------------------------

typedef __attribute__((ext_vector_type(2))) float v2f;
typedef __attribute__((ext_vector_type(4))) float v4f;
typedef __attribute__((ext_vector_type(8))) float v8f;

__constant__ int d_PARENT[14] = {13,13,13,13,13,13,13,0,0,0,0,0,0,0};
__constant__ int d_CHILD[14]  = {15,17,18,19,25,26,27,13,1,2,3,6,7,8};

#define HW        65536      // 256*256 pixels per channel
#define NTASK     59         // 17 L2D channels + 42 T channels
#define BLK_TASK  8          // blocks per task
#define WS_ACC    0          // ws[0..58]  : per-task sum-of-squares
#define WS_L3D    59         // ws[59]     : l3D
#define WS_R      60         // ws[60..73] : r per pair (-1/0/1 as float)

// ds_swizzle group-of-32 encoding: offset = (xor<<10)|(or<<5)|and.
// SWAPX16: xor=0x10, and=0x1f -> 0x401F (swap lane halves of the wave).
#define SWIZZLE_SWAPX16 0x401F

// Wave32 sum via V_WMMA_F32_16X16X4_F32:
//   A(16x4): lane L holds A[L&15][k] = acc in one K slot, 0 in the other.
//   B(4x16): all ones  =>  D[m][n] = acc[m] + acc[m+16]  (any n).
//   Lane L sums its 8 D VGPRs (rows 0-7 or 8-15), then one SWAPX16 swizzle
//   adds the other half's rows -> full wave sum in every lane.
// EXEC must be all ones at the call site (full blocks, no early returns).
__device__ __forceinline__ float wave_reduce_wmma(float acc) {
    v2f a; a.x = acc;  a.y = 0.0f;
    v2f b; b.x = 1.0f; b.y = 1.0f;
    v8f c = {};
    v8f d = __builtin_amdgcn_wmma_f32_16x16x4_f32(
        /*neg_a=*/false, a, /*neg_b=*/false, b,
        /*c_mod=*/(short)0, c, /*reuse_a=*/false, /*reuse_b=*/false);
    float s = ((d[0] + d[1]) + (d[2] + d[3])) + ((d[4] + d[5]) + (d[6] + d[7]));
    s += __int_as_float(__builtin_amdgcn_ds_swizzle(__float_as_int(s),
                                                    SWIZZLE_SWAPX16));
    return s;
}

__global__ void init_ws(float* ws) {
    int i = threadIdx.x;
    if (i < 80) ws[i] = 0.0f;
}

// One block: l3D (672-elem L1 reduction) + r[14] from joint3d z depths.
__global__ void small_kernel(const float* __restrict__ pred,
                             const float* __restrict__ joint3d,
                             const int*   __restrict__ corr,
                             float*       __restrict__ ws) {
    const int tid = threadIdx.x;
    if (tid < 14) {
        const int hp_i = corr[d_PARENT[tid]];
        const int hc_i = corr[d_CHILD[tid]];
        const float zp = joint3d[hp_i * 3 + 2];   // batch 0
        const float zc = joint3d[hc_i * 3 + 2];
        const float dz = zp - zc;
        float r = (dz > 0.1f) ? 1.0f : ((fabsf(dz) < 0.1f) ? 0.0f : -1.0f);
        ws[WS_R + tid] = r;
    }
    float acc = 0.0f;
    for (int i = tid; i < 8 * 28 * 3; i += blockDim.x)
        acc += fabsf(joint3d[i] - pred[i]);
    float s = wave_reduce_wmma(acc);               // EXEC all-ones here
    if ((tid & 31) == 0) atomicAdd(&ws[WS_L3D], s);
}

// Main streaming kernel: 59 tasks x 8 blocks x 256 threads.
// Unified per-pixel form:  acc += (c_hp*hp + c_hc*hc - mo)^2
//   task <  17 : L2D channel  (hp = heatmap[task], c_hp=1, c_hc=0)
//   task >= 17 : T channel    (p=(task-17)/3, j=(task-17)%3; T_pred chan == task)
__global__ void __launch_bounds__(256)
main_kernel(const float* __restrict__ middle_out,
            const float* __restrict__ heatmap,
            const int*   __restrict__ corr,
            float*       __restrict__ ws) {
    const int task = blockIdx.x >> 3;
    const int sub  = blockIdx.x & (BLK_TASK - 1);
    const int tid  = threadIdx.x;

    const float* mo = middle_out + (size_t)task * HW;   // batch 0, channel=task
    const float* hpp;
    const float* hcp;
    float c_hp, c_hc;

    if (task < 17) {
        hpp = heatmap + (size_t)task * HW;
        hcp = hpp;
        c_hp = 1.0f; c_hc = 0.0f;
    } else {
        const int u = task - 17;
        const int p = u / 3;
        const int j = u - p * 3;
        const int hp_i = corr[d_PARENT[p]];
        const int hc_i = corr[d_CHILD[p]];
        hpp = heatmap + (size_t)hp_i * HW;
        hcp = heatmap + (size_t)hc_i * HW;
        const float r = ws[WS_R + p];
        if (j == 0)      { c_hp = 0.0f; c_hc = (r == -1.0f) ? 1.0f : 0.0f; }
        else if (j == 1) { c_hp = 1.0f; c_hc = (r ==  0.0f) ? 1.0f : 0.0f; }
        else             { c_hp = 0.0f; c_hc = (r ==  1.0f) ? 1.0f : 0.0f; }
    }

    const v4f* mo4 = (const v4f*)mo;
    const v4f* hp4 = (const v4f*)hpp;
    const v4f* hc4 = (const v4f*)hcp;

    float acc = 0.0f;
    const int base = sub * (HW / 4 / BLK_TASK) + tid;   // 2048 float4 per block
#pragma unroll
    for (int k = 0; k < 8; ++k) {
        const int i = base + k * 256;                   // coalesced b128 stream
        const v4f m = __builtin_nontemporal_load(&mo4[i]);  // read-once -> NT
        const v4f p = hp4[i];                               // L2-resident
        const v4f q = hc4[i];
        float d0 = fmaf(c_hp, p[0], fmaf(c_hc, q[0], -m[0]));
        float d1 = fmaf(c_hp, p[1], fmaf(c_hc, q[1], -m[1]));
        float d2 = fmaf(c_hp, p[2], fmaf(c_hc, q[2], -m[2]));
        float d3 = fmaf(c_hp, p[3], fmaf(c_hc, q[3], -m[3]));
        acc = fmaf(d0, d0, acc);
        acc = fmaf(d1, d1, acc);
        acc = fmaf(d2, d2, acc);
        acc = fmaf(d3, d3, acc);
    }

    float s = wave_reduce_wmma(acc);   // WMMA wave reduction, EXEC all-ones
    if ((tid & 31) == 0) atomicAdd(&ws[WS_ACC + task], s);  // 8 atomics/block
}

// Tiny finalize: L2D = sum ws[0..16]; LHEM = sum_p (sum_j sqrt(ss))^2.
__global__ void final_kernel(const float* __restrict__ ws,
                             float*       __restrict__ out) {
    if (threadIdx.x == 0 && blockIdx.x == 0) {
        float L2D = 0.0f;
        for (int i = 0; i < 17; ++i) L2D += ws[i];
        float LHEM = 0.0f;
        for (int p = 0; p < 14; ++p) {
            float s = sqrtf(ws[17 + 3 * p + 0])
                    + sqrtf(ws[17 + 3 * p + 1])
                    + sqrtf(ws[17 + 3 * p + 2]);
            LHEM = fmaf(s, s, LHEM);
        }
        out[0] = ws[WS_L3D] + 0.05f * (LHEM + L2D);
    }
}

extern "C" void kernel_launch(void* const* d_in, const int* in_sizes, int n_in,
                              void* d_out, int out_size, void* d_ws, size_t ws_size,
                              hipStream_t stream) {
    const float* pred       = (const float*)d_in[0];   // (8,28,3)
    const float* joint3d    = (const float*)d_in[1];   // (8,28,3)
    const float* middle_out = (const float*)d_in[2];   // (8,64,256,256)
    const float* heatmap    = (const float*)d_in[3];   // (17,256,256)
    const int*   corr       = (const int*)d_in[4];     // (28,)
    float* ws  = (float*)d_ws;
    float* out = (float*)d_out;

    init_ws<<<1, 128, 0, stream>>>(ws);
    small_kernel<<<1, 256, 0, stream>>>(pred, joint3d, corr, ws);
    main_kernel<<<NTASK * BLK_TASK, 256, 0, stream>>>(middle_out, heatmap, corr, ws);
    final_kernel<<<1, 32, 0, stream>>>(ws, out);
}